// MatrixFactorization_10393820857075
// MI455X (gfx1250) — compile-verified
//
#include <hip/hip_runtime.h>
#include <hip/hip_bf16.h>

typedef float v2f __attribute__((ext_vector_type(2)));
typedef float v8f __attribute__((ext_vector_type(8)));

// LDS layout (in floats) for the four small embedding tables, each row = 32 floats
#define LDS_DOW   0       // 7*32  = 224
#define LDS_TIME  224     // 24*32 = 768
#define LDS_MONTH 992     // 12*32 = 384
#define LDS_DAY   1376    // 31*32 = 992
#define LDS_TOTAL 2368    // 9472 bytes

__global__ __launch_bounds__(256) void mf_rowdot_wmma_kernel(
    const int* __restrict__ dayofweek, const int* __restrict__ time_,
    const int* __restrict__ month,     const int* __restrict__ day,
    const int* __restrict__ destination,
    const float* __restrict__ W_dow,   const float* __restrict__ W_time,
    const float* __restrict__ W_month, const float* __restrict__ W_day,
    const float* __restrict__ W_item,
    float* __restrict__ out, int n, int total_tiles)
{
    __shared__ float smem[LDS_TOTAL];
    // Stage tiny tables into LDS (they are hot across every tile this block runs)
    for (int i = threadIdx.x; i < 224; i += blockDim.x) smem[LDS_DOW   + i] = W_dow[i];
    for (int i = threadIdx.x; i < 768; i += blockDim.x) smem[LDS_TIME  + i] = W_time[i];
    for (int i = threadIdx.x; i < 384; i += blockDim.x) smem[LDS_MONTH + i] = W_month[i];
    for (int i = threadIdx.x; i < 992; i += blockDim.x) smem[LDS_DAY   + i] = W_day[i];
    __syncthreads();

    const int lane          = threadIdx.x & 31;
    const int wave          = threadIdx.x >> 5;
    const int wavesPerBlock = blockDim.x >> 5;
    const int gwid          = blockIdx.x * wavesPerBlock + wave;
    const int nwaves        = gridDim.x * wavesPerBlock;

    // WMMA f32 16x16x4 fragment geometry (wave32):
    //  A frag: lane holds (M = lane&15, K = k0 + koff, k0 + koff + 1), koff = 2*(lane>>4)
    //  B frag: lane holds (N = lane&15, same K pair)  -> both are contiguous float2 loads
    const int nn   = lane & 15;          // sample-in-tile for B / M row for A
    const int koff = (lane >> 4) << 1;   // 0 or 2

    for (int t = gwid; t < total_tiles; t += nwaves) {
        int s = t * 16 + nn;
        if (s >= n) s = n - 1;  // clamp (keeps EXEC full through the WMMAs)

        // Streaming index loads: non-temporal so they don't evict the L2-resident W_item
        const int d  = __builtin_nontemporal_load(&destination[s]);
        const int i0 = __builtin_nontemporal_load(&dayofweek[s]);
        const int i1 = __builtin_nontemporal_load(&time_[s]);
        const int i2 = __builtin_nontemporal_load(&month[s]);
        const int i3 = __builtin_nontemporal_load(&day[s]);

        const float* bptr = W_item + (long)d * 128 + koff;   // B: item row, K-pair slice
        const float* ap[4];
        ap[0] = smem + LDS_DOW   + i0 * 32 + koff;           // A: user = concat of 4 tables
        ap[1] = smem + LDS_TIME  + i1 * 32 + koff;
        ap[2] = smem + LDS_MONTH + i2 * 32 + koff;
        ap[3] = smem + LDS_DAY   + i3 * 32 + koff;

        v8f c = {0.f, 0.f, 0.f, 0.f, 0.f, 0.f, 0.f, 0.f};
        #pragma unroll
        for (int f = 0; f < 4; ++f) {
            const float* a_f = ap[f];
            const float* b_f = bptr + f * 32;
            #pragma unroll
            for (int kl = 0; kl < 32; kl += 4) {
                v2f a = *(const v2f*)(a_f + kl);   // ds_load_b64
                v2f b = *(const v2f*)(b_f + kl);   // global_load_b64 (L2-resident)
                // D = A*B + C, f32 exact: v_wmma_f32_16x16x4_f32
                c = __builtin_amdgcn_wmma_f32_16x16x4_f32(
                        /*neg_a=*/false, a, /*neg_b=*/false, b,
                        /*c_mod=*/(short)0, c, /*reuse_a=*/false, /*reuse_b=*/false);
            }
        }

        // Extract diag(D): lanes 0..7 hold (m=lane) in c[lane];
        // lanes 24..31 hold (m=lane-16) in c[lane-24].
        const int r = lane & 7;
        float diag = c[0];
        diag = (r == 1) ? c[1] : diag;
        diag = (r == 2) ? c[2] : diag;
        diag = (r == 3) ? c[3] : diag;
        diag = (r == 4) ? c[4] : diag;
        diag = (r == 5) ? c[5] : diag;
        diag = (r == 6) ? c[6] : diag;
        diag = (r == 7) ? c[7] : diag;

        const bool writer = (lane < 8) || (lane >= 24);
        const int  m      = (lane < 8) ? lane : (lane - 16);
        const int  o      = t * 16 + m;
        if (writer && o < n) {
            __builtin_nontemporal_store(diag, &out[o]);  // streaming store, don't pollute L2
        }
    }
}

extern "C" void kernel_launch(void* const* d_in, const int* in_sizes, int n_in,
                              void* d_out, int out_size, void* d_ws, size_t ws_size,
                              hipStream_t stream) {
    const int*   dayofweek   = (const int*)  d_in[0];
    const int*   time_       = (const int*)  d_in[1];
    const int*   month       = (const int*)  d_in[2];
    const int*   day         = (const int*)  d_in[3];
    const int*   destination = (const int*)  d_in[4];
    const float* W_dow       = (const float*)d_in[5];
    const float* W_time      = (const float*)d_in[6];
    const float* W_month     = (const float*)d_in[7];
    const float* W_day       = (const float*)d_in[8];
    const float* W_item      = (const float*)d_in[9];
    float*       out         = (float*)d_out;

    const int n = in_sizes[0];
    if (n <= 0) return;

    const int total_tiles   = (n + 15) / 16;
    const int threads       = 256;                 // 8 waves per block (wave32)
    const int wavesPerBlock = threads / 32;
    const int tilesPerWave  = 4;                   // amortize LDS staging
    int blocks = (total_tiles + wavesPerBlock * tilesPerWave - 1) /
                 (wavesPerBlock * tilesPerWave);
    if (blocks < 1) blocks = 1;

    mf_rowdot_wmma_kernel<<<blocks, threads, 0, stream>>>(
        dayofweek, time_, month, day, destination,
        W_dow, W_time, W_month, W_day, W_item,
        out, n, total_tiles);
}